// GAT_86543591015075
// MI455X (gfx1250) — compile-verified
//
#include <hip/hip_runtime.h>

#define NEG_SLOPE 0.2f
#define HD 64

typedef float v2f __attribute__((ext_vector_type(2)));
typedef float v8f __attribute__((ext_vector_type(8)));

// ---- order-preserving float<->uint for atomic max on floats ----
__device__ __forceinline__ unsigned f2ord(float f) {
    unsigned u = __float_as_uint(f);
    return (u & 0x80000000u) ? ~u : (u | 0x80000000u);
}
__device__ __forceinline__ float ord2f(unsigned u) {
    return (u & 0x80000000u) ? __uint_as_float(u & 0x7fffffffu)
                             : __uint_as_float(~u);
}

__device__ __forceinline__ void edge_sd(const long long* adj, int e, int nE,
                                        int& s, int& d) {
    if (e < nE) { s = (int)adj[e]; d = (int)adj[nE + e]; }
    else        { s = d = e - nE; }
}

// ============ layer-1 linear: xl = x@Wl + bl, xr = x@Wr + br (K=3) ============
__global__ void lin1_kernel(const float* __restrict__ x,
                            const float* __restrict__ Wl, const float* __restrict__ bl,
                            const float* __restrict__ Wr, const float* __restrict__ br,
                            float* __restrict__ xl, float* __restrict__ xr, int n) {
    int i = blockIdx.x * blockDim.x + threadIdx.x;
    if (i >= n * HD) return;
    int node = i >> 6, h = i & 63;
    float x0 = x[node * 3 + 0], x1 = x[node * 3 + 1], x2 = x[node * 3 + 2];
    xl[i] = fmaf(x0, Wl[0 * HD + h], fmaf(x1, Wl[1 * HD + h], fmaf(x2, Wl[2 * HD + h], bl[h])));
    xr[i] = fmaf(x0, Wr[0 * HD + h], fmaf(x1, Wr[1 * HD + h], fmaf(x2, Wr[2 * HD + h], br[h])));
}

// ============ init segment-max / segment-sum / aggregation buffers ============
__global__ void init_kernel(unsigned* __restrict__ m, float* __restrict__ s,
                            float* __restrict__ agg, int n) {
    int i = blockIdx.x * blockDim.x + threadIdx.x;
    if (i < n) { m[i] = 0u; s[i] = 0.0f; }   // 0u < f2ord(-inf)... below all reals
    if (i < n * HD) agg[i] = 0.0f;
}

// ============ pass 1: e = att . leaky_relu(xl[src]+xr[dst]); seg-max ============
// wave-cooperative: 32 lanes cover 64 channels (2 each), coalesced row reads
__global__ void edge_attn_max_kernel(const long long* __restrict__ adj, int nE, int nTot,
                                     const float* __restrict__ xl, const float* __restrict__ xr,
                                     const float* __restrict__ att,
                                     float* __restrict__ ebuf, unsigned* __restrict__ m) {
    int lane = threadIdx.x & 31;
    int wave = (blockIdx.x * blockDim.x + threadIdx.x) >> 5;
    int nWaves = (gridDim.x * blockDim.x) >> 5;
    float a0 = att[lane], a1 = att[lane + 32];
    for (int e = wave; e < nTot; e += nWaves) {
        int s, d; edge_sd(adj, e, nE, s, d);
        const float* ps = xl + (size_t)s * HD;
        const float* pd = xr + (size_t)d * HD;
        float v0 = ps[lane] + pd[lane];
        float v1 = ps[lane + 32] + pd[lane + 32];
        v0 = v0 > 0.0f ? v0 : NEG_SLOPE * v0;
        v1 = v1 > 0.0f ? v1 : NEG_SLOPE * v1;
        float p = fmaf(a0, v0, a1 * v1);
        #pragma unroll
        for (int off = 16; off > 0; off >>= 1) p += __shfl_xor(p, off, 32);
        if (lane == 0) {
            ebuf[e] = p;
            atomicMax(&m[d], f2ord(p));
        }
    }
}

// ============ pass 2: ex = exp(e - m[dst]); seg-sum ============
__global__ void edge_expsum_kernel(const long long* __restrict__ adj, int nE, int nTot,
                                   const unsigned* __restrict__ m,
                                   float* __restrict__ ebuf, float* __restrict__ s) {
    int e = blockIdx.x * blockDim.x + threadIdx.x;
    if (e >= nTot) return;
    int si, d; edge_sd(adj, e, nE, si, d);
    float ex = __expf(ebuf[e] - ord2f(m[d]));
    ebuf[e] = ex;
    atomicAdd(&s[d], ex);
}

// ============ pass 3: agg[dst] += (ex/s[dst]) * xl[src] ============
__global__ void edge_scatter_kernel(const long long* __restrict__ adj, int nE, int nTot,
                                    const float* __restrict__ ebuf, const float* __restrict__ s,
                                    const float* __restrict__ xl, float* __restrict__ agg) {
    int lane = threadIdx.x & 31;
    int wave = (blockIdx.x * blockDim.x + threadIdx.x) >> 5;
    int nWaves = (gridDim.x * blockDim.x) >> 5;
    for (int e = wave; e < nTot; e += nWaves) {
        int sn, d; edge_sd(adj, e, nE, sn, d);
        float alpha = ebuf[e] / s[d];
        const float* ps = xl + (size_t)sn * HD;
        float* pa = agg + (size_t)d * HD;
        atomicAdd(&pa[lane], alpha * ps[lane]);
        atomicAdd(&pa[lane + 32], alpha * ps[lane + 32]);
    }
}

// ============ h = relu(agg + bias) ============
__global__ void bias_relu_kernel(const float* __restrict__ agg, const float* __restrict__ bias,
                                 float* __restrict__ out, int n) {
    int i = blockIdx.x * blockDim.x + threadIdx.x;
    if (i >= n * HD) return;
    float v = agg[i] + bias[i & 63];
    out[i] = v > 0.0f ? v : 0.0f;
}

// ============ layer-2 linear via WMMA fp32: xl2=h1@Wl+bl, xr2=h1@Wr+br ============
// Block: 256 threads = 8 waves. Waves 0-3 -> Wl col-tiles 0-3; waves 4-7 -> Wr.
// Each wave: one 16x16 output tile, K=64 in 16 steps of V_WMMA_F32_16X16X4_F32.
__global__ __launch_bounds__(256) void lin2_wmma_kernel(
        const float* __restrict__ Hm,
        const float* __restrict__ Wl, const float* __restrict__ bl,
        const float* __restrict__ Wr, const float* __restrict__ br,
        float* __restrict__ xl, float* __restrict__ xr) {
    int wave = threadIdx.x >> 5;
    int lane = threadIdx.x & 31;
    const float* W   = (wave < 4) ? Wl : Wr;
    const float* bia = (wave < 4) ? bl : br;
    float*       out = (wave < 4) ? xl : xr;
    int ct = (wave & 3) * 16;          // column tile base
    int half = lane >> 4;              // 0: lanes 0-15, 1: lanes 16-31
    int lr = lane & 15;
    int rowBase = blockIdx.x * 16;     // N = 50000 = 3125 * 16, exact

    v8f c = {};
    #pragma unroll
    for (int k = 0; k < 16; ++k) {
        int kb = k * 4 + half * 2;     // this half-wave's K pair
        v2f a, b;
        // A 16x4 fragment: lane lr holds row rowBase+lr, K = kb, kb+1
        const float* ap = Hm + (size_t)(rowBase + lr) * HD + kb;
        a.x = ap[0]; a.y = ap[1];
        // B 4x16 fragment: lane lr holds col ct+lr, K rows kb, kb+1
        const float* bp = W + (size_t)kb * HD + ct + lr;
        b.x = bp[0]; b.y = bp[HD];
        c = __builtin_amdgcn_wmma_f32_16x16x4_f32(
                /*neg_a=*/false, a, /*neg_b=*/false, b,
                /*c_mod=*/(short)0, c, /*reuse_a=*/false, /*reuse_b=*/false);
    }
    // C/D layout: VGPR r -> M = r (lanes 0-15) / r+8 (lanes 16-31), N = lane&15
    int col = ct + lr;
    float bv = bia[col];
    #pragma unroll
    for (int r = 0; r < 8; ++r) {
        int row = rowBase + r + half * 8;
        out[(size_t)row * HD + col] = c[r] + bv;
    }
}

// ============ out = h2 @ W_fc + b_fc (64 -> 1), wave per node ============
__global__ void fc_kernel(const float* __restrict__ h2, const float* __restrict__ Wfc,
                          const float* __restrict__ bfc, float* __restrict__ out, int n) {
    int lane = threadIdx.x & 31;
    int node = (blockIdx.x * blockDim.x + threadIdx.x) >> 5;
    if (node >= n) return;
    const float* p = h2 + (size_t)node * HD;
    float acc = fmaf(p[lane], Wfc[lane], p[lane + 32] * Wfc[lane + 32]);
    #pragma unroll
    for (int off = 16; off > 0; off >>= 1) acc += __shfl_xor(acc, off, 32);
    if (lane == 0) out[node] = acc + bfc[0];
}

extern "C" void kernel_launch(void* const* d_in, const int* in_sizes, int n_in,
                              void* d_out, int out_size, void* d_ws, size_t ws_size,
                              hipStream_t stream) {
    (void)n_in; (void)ws_size;
    const float*     x    = (const float*)d_in[0];
    // d_in[1] = c (unused by reference), d_in[3] = edge_w (unused)
    const long long* adj  = (const long long*)d_in[2];
    const float* W1l = (const float*)d_in[4],  *b1l = (const float*)d_in[5];
    const float* W1r = (const float*)d_in[6],  *b1r = (const float*)d_in[7];
    const float* att1 = (const float*)d_in[8], *bias1 = (const float*)d_in[9];
    const float* W2l = (const float*)d_in[10], *b2l = (const float*)d_in[11];
    const float* W2r = (const float*)d_in[12], *b2r = (const float*)d_in[13];
    const float* att2 = (const float*)d_in[14], *bias2 = (const float*)d_in[15];
    const float* Wfc = (const float*)d_in[16], *bfc = (const float*)d_in[17];

    const int n    = in_sizes[0] / 3;      // 50000
    const int E    = in_sizes[2] / 2;      // 1600000
    const int nTot = E + n;                // edges incl. self-loops

    // workspace layout (floats)
    float* ws   = (float*)d_ws;
    float* xl   = ws;                      // n*64
    float* xr   = xl + (size_t)n * HD;     // n*64
    float* agg  = xr + (size_t)n * HD;     // n*64
    float* h1   = agg + (size_t)n * HD;    // n*64 (later reused for h2)
    float* ebuf = h1 + (size_t)n * HD;     // nTot
    unsigned* m = (unsigned*)(ebuf + nTot);// n
    float* sseg = (float*)(m + n);         // n

    const int T = 256;
    dim3 gNH((n * HD + T - 1) / T), gT(T);
    dim3 gEdgeThread((nTot + T - 1) / T);
    dim3 gEdgeWave(2048);                  // grid-stride, 16384 waves
    dim3 gWmma(n / 16);                    // 3125 tiles, exact
    dim3 gFc((n * 32 + T - 1) / T);

    // ---------- layer 1 ----------
    lin1_kernel<<<gNH, gT, 0, stream>>>(x, W1l, b1l, W1r, b1r, xl, xr, n);
    init_kernel<<<gNH, gT, 0, stream>>>(m, sseg, agg, n);
    edge_attn_max_kernel<<<gEdgeWave, gT, 0, stream>>>(adj, E, nTot, xl, xr, att1, ebuf, m);
    edge_expsum_kernel<<<gEdgeThread, gT, 0, stream>>>(adj, E, nTot, m, ebuf, sseg);
    edge_scatter_kernel<<<gEdgeWave, gT, 0, stream>>>(adj, E, nTot, ebuf, sseg, xl, agg);
    bias_relu_kernel<<<gNH, gT, 0, stream>>>(agg, bias1, h1, n);

    // ---------- layer 2 ----------
    lin2_wmma_kernel<<<gWmma, gT, 0, stream>>>(h1, W2l, b2l, W2r, b2r, xl, xr);
    init_kernel<<<gNH, gT, 0, stream>>>(m, sseg, agg, n);
    edge_attn_max_kernel<<<gEdgeWave, gT, 0, stream>>>(adj, E, nTot, xl, xr, att2, ebuf, m);
    edge_expsum_kernel<<<gEdgeThread, gT, 0, stream>>>(adj, E, nTot, m, ebuf, sseg);
    edge_scatter_kernel<<<gEdgeWave, gT, 0, stream>>>(adj, E, nTot, ebuf, sseg, xl, agg);
    bias_relu_kernel<<<gNH, gT, 0, stream>>>(agg, bias2, h1, n);   // h1 now holds h2

    // ---------- final FC ----------
    fc_kernel<<<gFc, gT, 0, stream>>>(h1, Wfc, bfc, (float*)d_out, n);
}